// DecoderRNN_10428180595409
// MI455X (gfx1250) — compile-verified
//
#include <hip/hip_runtime.h>
#include <hip/hip_bf16.h>

// Problem constants (match reference): B,T,P,F,E,H,V
#define NB 64
#define NT 20
#define NP 49
#define NF 2048
#define NE 256
#define NH 512
#define NV 30000

typedef __attribute__((ext_vector_type(2))) float v2f;
typedef __attribute__((ext_vector_type(8))) float v8f;

// ---------------------------------------------------------------------------
// fp32 WMMA GEMM:  Cout[M,N] = A[M,K] @ B[K,N] + bias[N] (+ Cin[M,N])
// Each wave32 computes a 16 x (16*NTile) strip: NTile accumulators share one
// A fragment per k-step.  Copy-free ping-pong double buffering with a fully
// branch-free steady-state loop body (no joins -> no conservative
// s_wait_loadcnt): each WMMA group waits only on loads issued one full WMMA
// group earlier.  Final two groups are peeled into an epilogue.
// Requires M%16==0, N%(16*NTile)==0, K%8==0, K>=16.  All branches
// wave-uniform (EXEC all-1s for every WMMA, as the ISA requires).
// ---------------------------------------------------------------------------
template <int NTile>
struct Frag {
    v2f a;
    v2f b[NTile];
};

template <int NTile>
__global__ void __launch_bounds__(256, 1)
wmma_gemm_f32(const float* __restrict__ A, const float* __restrict__ B,
              const float* __restrict__ bias, const float* __restrict__ Cin,
              float* __restrict__ Cout,
              int M, int N, int K, int lda, int ldb, int ldc)
{
    const int wave   = blockIdx.x * (blockDim.x >> 5) + (threadIdx.x >> 5);
    const int groups = N / (16 * NTile);
    const int tiles  = (M >> 4) * groups;
    if (wave >= tiles) return;               // wave-uniform

    const int m0   = (wave / groups) << 4;
    const int n0   = (wave % groups) * (16 * NTile);
    const int lane = threadIdx.x & 31;
    const int half = lane >> 4;              // 0: lanes 0-15, 1: lanes 16-31
    const int r    = lane & 15;

    // A frag: VGPR v = A[m0+r][kk + v + 2*half]   (contiguous 8B pair)
    // B frag i: VGPR v = B[kk + v + 2*half][n0 + 16*i + r]
    const float* __restrict__ Ap = A + (size_t)(m0 + r) * lda + (half << 1);
    const float* __restrict__ Bp = B + (size_t)(half << 1) * ldb + n0 + r;

    auto load_frag = [&](int kk) {
        Frag<NTile> f;
        f.a.x = Ap[kk];
        f.a.y = Ap[kk + 1];
        #pragma unroll
        for (int i = 0; i < NTile; ++i) {
            f.b[i].x = Bp[(size_t)kk * ldb + 16 * i];
            f.b[i].y = Bp[(size_t)(kk + 1) * ldb + 16 * i];
        }
        return f;
    };

    v8f acc[NTile];
    #pragma unroll
    for (int i = 0; i < NTile; ++i) acc[i] = (v8f){};

    auto mma_group = [&](const Frag<NTile>& f) {
        #pragma unroll
        for (int i = 0; i < NTile; ++i) {
            acc[i] = __builtin_amdgcn_wmma_f32_16x16x4_f32(
                         /*neg_a=*/false, f.a, /*neg_b=*/false, f.b[i],
                         /*c_mod=*/(short)0, acc[i],
                         /*reuse_a=*/false, /*reuse_b=*/false);
        }
    };

    constexpr int PF = 32;                   // prefetch distance (k elements)

    Frag<NTile> f0 = load_frag(0);
    Frag<NTile> f1 = load_frag(4);

    // Branch-free steady state: loads are unconditional; prefetch row index
    // is clamped instead of guarded (re-prefetching a fetched row is benign).
    int kk = 0;
    for (; kk + 16 <= K; kk += 8) {
        int kpf = kk + PF;
        if (kpf > K - 4) kpf = K - 4;        // folds to s_min (branchless)
        __builtin_prefetch(Bp + (size_t)kpf * ldb, 0, 3);
        mma_group(f0);
        f0 = load_frag(kk + 8);              // covered by group(f1)
        mma_group(f1);
        f1 = load_frag(kk + 12);             // covered by next group(f0)
    }
    // kk == K-8 here (K%8==0): consume the last two fragments, no loads left.
    mma_group(f0);
    mma_group(f1);

    #pragma unroll
    for (int i = 0; i < NTile; ++i) {
        const int   nc = n0 + 16 * i + r;
        const float bv = bias ? bias[nc] : 0.0f;
        #pragma unroll
        for (int j = 0; j < 8; ++j) {
            const int    row = m0 + j + (half << 3);  // C/D: VGPR j -> M=j+8*half
            const size_t off = (size_t)row * ldc + nc;
            float v = acc[i][j] + bv;
            if (Cin) v += Cin[off];
            Cout[off] = v;
        }
    }
}

// ---------------------------------------------------------------------------
// Additive attention: per batch b, e_p = v . relu(enc_proj[b,p,:] + dec[b,:]),
// alpha = softmax_p(e).  (b_att is a uniform shift -> softmax-invariant.)
// One block (8 waves) per batch element.
// ---------------------------------------------------------------------------
__global__ void __launch_bounds__(256)
attention_kernel(const float* __restrict__ enc_proj, const float* __restrict__ dec,
                 const float* __restrict__ v_att, float* __restrict__ alpha)
{
    __shared__ float s_dec[NH];
    __shared__ float s_e[64];
    __shared__ float s_inv;

    const int b    = blockIdx.x;
    const int tid  = threadIdx.x;
    const int wv   = tid >> 5;
    const int lane = tid & 31;

    for (int i = tid; i < NH; i += 256) s_dec[i] = dec[b * NH + i];
    __syncthreads();

    for (int p = wv; p < NP; p += 8) {
        const float* ep = enc_proj + ((size_t)b * NP + p) * NH;
        float s = 0.0f;
        for (int h = lane; h < NH; h += 32) {
            float t = ep[h] + s_dec[h];
            s += fmaxf(t, 0.0f) * v_att[h];
        }
        #pragma unroll
        for (int off = 16; off > 0; off >>= 1) s += __shfl_xor(s, off, 32);
        if (lane == 0) s_e[p] = s;
    }
    __syncthreads();

    if (tid == 0) {
        float m = s_e[0];
        for (int p = 1; p < NP; ++p) m = fmaxf(m, s_e[p]);
        float s = 0.0f;
        for (int p = 0; p < NP; ++p) { float e = __expf(s_e[p] - m); s_e[p] = e; s += e; }
        s_inv = 1.0f / s;
    }
    __syncthreads();
    if (tid < NP) alpha[b * NP + tid] = s_e[tid] * s_inv;
}

// ---------------------------------------------------------------------------
// Build LSTM input x[b, 0:E] = emb_table[captions[b,t]], x[b, E:E+F] = context
// where context[b,f] = sum_p alpha[b,p] * features[b,p,f].
// grid = (B, (E+F)/256)
// ---------------------------------------------------------------------------
__global__ void __launch_bounds__(256)
ctx_x_kernel(const float* __restrict__ features, const float* __restrict__ alpha,
             const float* __restrict__ emb_table, const int* __restrict__ captions,
             float* __restrict__ x, int t)
{
    __shared__ float s_alpha[NP];
    const int b = blockIdx.x;
    const int j = blockIdx.y * 256 + threadIdx.x;      // 0 .. E+F-1
    if (threadIdx.x < NP) s_alpha[threadIdx.x] = alpha[b * NP + threadIdx.x];
    __syncthreads();

    float val;
    if (j < NE) {
        const int tok = captions[b * NT + t];
        val = emb_table[(size_t)tok * NE + j];
    } else {
        const int f = j - NE;
        const float* fb = features + ((size_t)b * NP) * NF + f;
        float s = 0.0f;
        #pragma unroll 7
        for (int p = 0; p < NP; ++p) s += s_alpha[p] * fb[(size_t)p * NF];
        val = s;
    }
    x[(size_t)b * (NE + NF) + j] = val;
}

// ---------------------------------------------------------------------------
// LSTM elementwise: gates laid out [B, 4H] in PyTorch (i,f,g,o) chunk order.
// ---------------------------------------------------------------------------
__global__ void __launch_bounds__(256)
lstm_kernel(const float* __restrict__ gates, float* __restrict__ h, float* __restrict__ c)
{
    const int idx = blockIdx.x * 256 + threadIdx.x;    // 0 .. B*H-1
    if (idx >= NB * NH) return;
    const int b  = idx >> 9;
    const int hh = idx & (NH - 1);
    const float* g = gates + (size_t)b * 4 * NH;
    const float gi = g[hh];
    const float gf = g[NH + hh];
    const float gg = g[2 * NH + hh];
    const float go = g[3 * NH + hh];
    const float si = 1.0f / (1.0f + __expf(-gi));
    const float sf = 1.0f / (1.0f + __expf(-gf));
    const float so = 1.0f / (1.0f + __expf(-go));
    const float c2 = sf * c[idx] + si * tanhf(gg);
    c[idx] = c2;
    h[idx] = so * tanhf(c2);
}

__global__ void __launch_bounds__(256)
zero_kernel(float* __restrict__ p, int n)
{
    const int i = blockIdx.x * 256 + threadIdx.x;
    if (i < n) p[i] = 0.0f;
}

static inline int gemm_blocks(int M, int N, int ntile)
{
    const int tiles = (M >> 4) * (N / (16 * ntile));
    return (tiles + 7) / 8;
}

extern "C" void kernel_launch(void* const* d_in, const int* in_sizes, int n_in,
                              void* d_out, int out_size, void* d_ws, size_t ws_size,
                              hipStream_t stream)
{
    const float* features = (const float*)d_in[0];   // [B,P,F]
    const int*   captions = (const int*)  d_in[1];   // [B,T]
    const float* emb      = (const float*)d_in[2];   // [V,E]
    const float* W_enc    = (const float*)d_in[3];   // [F,H]
    const float* b_enc    = (const float*)d_in[4];
    const float* W_dec    = (const float*)d_in[5];   // [H,H]
    const float* b_dec    = (const float*)d_in[6];
    const float* v_att    = (const float*)d_in[7];
    /* d_in[8] = b_att: uniform shift before softmax -> no effect, skipped */
    const float* W_ih     = (const float*)d_in[9];   // [E+F,4H]
    const float* b_ih     = (const float*)d_in[10];
    const float* W_hh     = (const float*)d_in[11];  // [H,4H]
    const float* b_hh     = (const float*)d_in[12];
    const float* W_fc     = (const float*)d_in[13];  // [H,V]
    const float* b_fc     = (const float*)d_in[14];
    float* out = (float*)d_out;                      // [B,T,V]

    // Workspace layout (floats): ~7.95 MB total
    float* ws       = (float*)d_ws;
    float* enc_proj = ws;                                  // B*P*H = 1,605,632
    float* h        = enc_proj + (size_t)NB * NP * NH;     // B*H
    float* c        = h        + (size_t)NB * NH;          // B*H
    float* dec      = c        + (size_t)NB * NH;          // B*H
    float* alpha    = dec      + (size_t)NB * NH;          // B*64 (padded)
    float* x        = alpha    + (size_t)NB * 64;          // B*(E+F)
    float* gates    = x        + (size_t)NB * (NE + NF);   // B*4H

    // h = c = 0
    zero_kernel<<<(2 * NB * NH + 255) / 256, 256, 0, stream>>>(h, 2 * NB * NH);

    // enc_proj[B*P, H] = features[B*P, F] @ W_enc[F, H] + b_enc   (hoisted)
    wmma_gemm_f32<4><<<gemm_blocks(NB * NP, NH, 4), 256, 0, stream>>>(
        features, W_enc, b_enc, nullptr, enc_proj,
        NB * NP, NH, NF, NF, NH, NH);

    for (int t = 0; t < NT; ++t) {
        // dec = h @ W_dec + b_dec
        wmma_gemm_f32<4><<<gemm_blocks(NB, NH, 4), 256, 0, stream>>>(
            h, W_dec, b_dec, nullptr, dec, NB, NH, NH, NH, NH, NH);

        // alpha = softmax_p( v . relu(enc_proj + dec) )
        attention_kernel<<<NB, 256, 0, stream>>>(enc_proj, dec, v_att, alpha);

        // x = [emb_t | alpha @ features]
        ctx_x_kernel<<<dim3(NB, (NE + NF) / 256), 256, 0, stream>>>(
            features, alpha, emb, captions, x, t);

        // gates = x @ W_ih + b_ih      then      gates += h @ W_hh + b_hh
        wmma_gemm_f32<4><<<gemm_blocks(NB, 4 * NH, 4), 256, 0, stream>>>(
            x, W_ih, b_ih, nullptr, gates, NB, 4 * NH, NE + NF, NE + NF, 4 * NH, 4 * NH);
        wmma_gemm_f32<4><<<gemm_blocks(NB, 4 * NH, 4), 256, 0, stream>>>(
            h, W_hh, b_hh, gates, gates, NB, 4 * NH, NH, NH, 4 * NH, 4 * NH);

        // (h, c) update
        lstm_kernel<<<(NB * NH + 255) / 256, 256, 0, stream>>>(gates, h, c);

        // logits: out[:, t, :] = h @ W_fc + b_fc   (row stride T*V)
        // 30000 = 375 * 80 -> NTile=5, exact, no predication.
        wmma_gemm_f32<5><<<gemm_blocks(NB, NV, 5), 256, 0, stream>>>(
            h, W_fc, b_fc, nullptr, out + (size_t)t * NV,
            NB, NV, NH, NH, NV, NT * NV);
    }
}